// network_representation_module_GCN_residual_57346403336483
// MI455X (gfx1250) — compile-verified
//
#include <hip/hip_runtime.h>
#include <hip/hip_bf16.h>

typedef __attribute__((ext_vector_type(2))) float v2f;
typedef __attribute__((ext_vector_type(8))) float v8f;

#define SCALE_F 0.70710678118654752440f
#define BN_EPS_F 1e-5f

// ---------------------------------------------------------------------------
// zero fill
__global__ void zero_f32_k(float* __restrict__ p, long n) {
    long i = (long)blockIdx.x * blockDim.x + threadIdx.x;
    if (i < n) p[i] = 0.0f;
}

// ---------------------------------------------------------------------------
// degree accumulation: deg_out[src[e]] += 1 ; deg_in[dst[e]] += 1
__global__ void degree_k(const int* __restrict__ src, const int* __restrict__ dst,
                         float* __restrict__ deg_out, float* __restrict__ deg_in, int E) {
    int e = blockIdx.x * blockDim.x + threadIdx.x;
    if (e < E) {
        atomicAdd(&deg_out[src[e]], 1.0f);
        atomicAdd(&deg_in[dst[e]], 1.0f);
    }
}

// in-place: v = rsqrt(max(v,1))   (applied to both deg arrays, 2N contiguous)
__global__ void norm_finalize_k(float* __restrict__ p, long n) {
    long i = (long)blockIdx.x * blockDim.x + threadIdx.x;
    if (i < n) p[i] = rsqrtf(fmaxf(p[i], 1.0f));
}

// ---------------------------------------------------------------------------
// C[M x 64] = A[M x K] @ W[K x 64] (+ bias) using V_WMMA_F32_16X16X4_F32.
// One wave handles 16 rows x 64 cols (4 accumulator tiles).
// Lane mapping (wave32): l = lane&15, half = lane>>4.
//  A (16x4):  vgpr0 = A[l][k0+2*half], vgpr1 = A[l][k0+2*half+1]
//  B (4x16):  vgpr0 = W[k0+2*half][col+l], vgpr1 = W[k0+2*half+1][col+l]
//  D (16x16): vgpr r = C[r+8*half][col+l]
__global__ void gemm_wmma_k(const float* __restrict__ A, const float* __restrict__ W,
                            const float* __restrict__ bias, float* __restrict__ C,
                            int M, int K) {
    const int lane = threadIdx.x & 31;
    const int wave = threadIdx.x >> 5;
    const int rb = blockIdx.x * (blockDim.x >> 5) + wave;
    if (rb * 16 >= M) return;               // wave-uniform: EXEC stays all-ones
    const int l = lane & 15;
    const int half = lane >> 4;

    const float* __restrict__ arow = A + (long)(rb * 16 + l) * K;

    v8f acc0 = {}, acc1 = {}, acc2 = {}, acc3 = {};
    for (int k0 = 0; k0 < K; k0 += 4) {
        const int ka = k0 + 2 * half;
        v2f a;  a.x = arow[ka];             a.y = arow[ka + 1];
        const float* __restrict__ w0 = W + (long)ka * 64;
        const float* __restrict__ w1 = w0 + 64;
        v2f b0; b0.x = w0[l];       b0.y = w1[l];
        v2f b1; b1.x = w0[16 + l];  b1.y = w1[16 + l];
        v2f b2; b2.x = w0[32 + l];  b2.y = w1[32 + l];
        v2f b3; b3.x = w0[48 + l];  b3.y = w1[48 + l];
        acc0 = __builtin_amdgcn_wmma_f32_16x16x4_f32(false, a, false, b0, (short)0, acc0, false, false);
        acc1 = __builtin_amdgcn_wmma_f32_16x16x4_f32(false, a, false, b1, (short)0, acc1, false, false);
        acc2 = __builtin_amdgcn_wmma_f32_16x16x4_f32(false, a, false, b2, (short)0, acc2, false, false);
        acc3 = __builtin_amdgcn_wmma_f32_16x16x4_f32(false, a, false, b3, (short)0, acc3, false, false);
    }

    float bb0 = 0.f, bb1 = 0.f, bb2 = 0.f, bb3 = 0.f;
    if (bias) { bb0 = bias[l]; bb1 = bias[16 + l]; bb2 = bias[32 + l]; bb3 = bias[48 + l]; }

#pragma unroll
    for (int r = 0; r < 8; ++r) {
        float* __restrict__ crow = C + (long)(rb * 16 + r + 8 * half) * 64;
        crow[l]      = acc0[r] + bb0;
        crow[16 + l] = acc1[r] + bb1;
        crow[32 + l] = acc2[r] + bb2;
        crow[48 + l] = acc3[r] + bb3;
    }
}

// ---------------------------------------------------------------------------
// per-edge gather + scatter-add: agg[dst] += hw[src] * norm_src[src]
// one wave per edge, 2 floats per lane (64 features)
__global__ void edge_scatter_k(const float* __restrict__ hw, const int* __restrict__ src,
                               const int* __restrict__ dst, const float* __restrict__ ns,
                               float* __restrict__ agg, int E) {
    long tid = (long)blockIdx.x * blockDim.x + threadIdx.x;
    int e = (int)(tid >> 5);
    if (e >= E) return;
    int lane = (int)(tid & 31);
    int s = src[e], d = dst[e];
    float nv = ns[s];
    const float2 v = *(const float2*)(hw + (long)s * 64 + lane * 2);
    float* ap = agg + (long)d * 64 + lane * 2;
    atomicAdd(ap,     v.x * nv);
    atomicAdd(ap + 1, v.y * nv);
}

// ---------------------------------------------------------------------------
// out = (h + agg * norm_dst[n] + b[f]) * sqrt(0.5)
__global__ void combine_k(const float* __restrict__ h, const float* __restrict__ agg,
                          const float* __restrict__ nd, const float* __restrict__ b,
                          float* __restrict__ out, long total) {
    long i = (long)blockIdx.x * blockDim.x + threadIdx.x;
    if (i < total) {
        long n = i >> 6;
        int f = (int)(i & 63);
        out[i] = (h[i] + agg[i] * nd[n] + b[f]) * SCALE_F;
    }
}

// ---------------------------------------------------------------------------
// column sums / sums-of-squares  (blockDim = 256 = 64 cols x 4 row groups)
__global__ void stats_k(const float* __restrict__ h, float* __restrict__ stats, int N) {
    int f = threadIdx.x & 63;
    int g = threadIdx.x >> 6;
    int stride = gridDim.x * 4;
    float s = 0.f, sq = 0.f;
    for (int n = blockIdx.x * 4 + g; n < N; n += stride) {
        float v = h[(long)n * 64 + f];
        s += v; sq += v * v;
    }
    __shared__ float ss[256];
    __shared__ float sv[256];
    ss[threadIdx.x] = s; sv[threadIdx.x] = sq;
    __syncthreads();
    if (g == 0) {
        s  = ss[f] + ss[64 + f] + ss[128 + f] + ss[192 + f];
        sq = sv[f] + sv[64 + f] + sv[128 + f] + sv[192 + f];
        atomicAdd(&stats[f], s);
        atomicAdd(&stats[64 + f], sq);
    }
}

// out = (out - mu) * rsqrt(var + eps) * gamma + beta   (biased var = E[x^2]-mu^2)
__global__ void bn_apply_k(float* __restrict__ out, const float* __restrict__ stats,
                           const float* __restrict__ gamma, const float* __restrict__ beta,
                           long total, float invN) {
    long i = (long)blockIdx.x * blockDim.x + threadIdx.x;
    if (i < total) {
        int f = (int)(i & 63);
        float mu  = stats[f] * invN;
        float var = stats[64 + f] * invN - mu * mu;
        float inv = rsqrtf(var + BN_EPS_F);
        out[i] = (out[i] - mu) * inv * gamma[f] + beta[f];
    }
}

// ---------------------------------------------------------------------------
extern "C" void kernel_launch(void* const* d_in, const int* in_sizes, int n_in,
                              void* d_out, int out_size, void* d_ws, size_t ws_size,
                              hipStream_t stream) {
    const int*   src   = (const int*)d_in[0];
    const int*   dst   = (const int*)d_in[1];
    const float* x     = (const float*)d_in[2];
    const float* fc_w  = (const float*)d_in[3];
    const float* fc_b  = (const float*)d_in[4];
    const float* w1    = (const float*)d_in[5];
    const float* b1    = (const float*)d_in[6];
    const float* w2    = (const float*)d_in[7];
    const float* b2    = (const float*)d_in[8];
    const float* gamma = (const float*)d_in[9];
    const float* beta  = (const float*)d_in[10];

    const int E = in_sizes[0];
    const int N = in_sizes[2] / 128;   // 100000
    const int K_IN = 128;
    const long total = (long)N * 64;

    float* ws    = (float*)d_ws;
    float* ns    = ws;                         // [N]  deg_out -> norm_src per node
    float* nd    = ws + N;                     // [N]  deg_in  -> norm_dst per node
    float* h     = ws + 2L * N;                // [N*64] running features
    float* hw    = h + total;                  // [N*64] h @ w  (pre-aggregation)
    float* stats = hw + total;                 // [128] col sums / sumsq
    float* agg   = (float*)d_out;              // [N*64] aggregation buffer (aliased)

    const int B = 256;
    const int gE   = (E + B - 1) / B;
    const int gTot = (int)((total + B - 1) / B);
    const int g2N  = (2 * N + B - 1) / B;
    const int gEdge = (int)(((long)E * 32 + B - 1) / B);
    const int rowBlocks = (N + 15) / 16;
    const int gGemm = (rowBlocks + 7) / 8;     // 8 waves per block

    // degrees -> norms
    zero_f32_k<<<g2N, B, 0, stream>>>(ns, 2L * N);
    zero_f32_k<<<1, 128, 0, stream>>>(stats, 128);
    degree_k<<<gE, B, 0, stream>>>(src, dst, ns, nd, E);
    norm_finalize_k<<<g2N, B, 0, stream>>>(ns, 2L * N);

    // fc: h = x @ fc_w + fc_b
    gemm_wmma_k<<<gGemm, B, 0, stream>>>(x, fc_w, fc_b, h, N, K_IN);

    // layer 1
    zero_f32_k<<<gTot, B, 0, stream>>>(agg, total);
    gemm_wmma_k<<<gGemm, B, 0, stream>>>(h, w1, nullptr, hw, N, 64);
    edge_scatter_k<<<gEdge, B, 0, stream>>>(hw, src, dst, ns, agg, E);
    combine_k<<<gTot, B, 0, stream>>>(h, agg, nd, b1, h, total);

    // layer 2 (pre-BN result into d_out; agg aliases d_out, elementwise-safe)
    zero_f32_k<<<gTot, B, 0, stream>>>(agg, total);
    gemm_wmma_k<<<gGemm, B, 0, stream>>>(h, w2, nullptr, hw, N, 64);
    edge_scatter_k<<<gEdge, B, 0, stream>>>(hw, src, dst, ns, agg, E);
    combine_k<<<gTot, B, 0, stream>>>(h, agg, nd, b2, (float*)d_out, total);

    // batch norm
    stats_k<<<128, B, 0, stream>>>((const float*)d_out, stats, N);
    bn_apply_k<<<gTot, B, 0, stream>>>((float*)d_out, stats, gamma, beta, total, 1.0f / (float)N);
}